// IndDipole_62440234549687
// MI455X (gfx1250) — compile-verified
//
#include <hip/hip_runtime.h>
#include <math.h>

// CDNA5 / gfx1250, wave32. WMMA f32 16x16x4 used for the node MLPs.
typedef __attribute__((ext_vector_type(2))) float v2f;
typedef __attribute__((ext_vector_type(8))) float v8f;

#define CH  64
#define CH2 128
#define LN_EPS 1e-5f

__device__ __forceinline__ float silu_f(float v) {
    return v * (1.0f / (1.0f + __expf(-v)));
}

// ---------------------------------------------------------------------------
// Node kernel: LayerNorm + p/q MLPs via V_WMMA_F32_16X16X4_F32.
// Block = 64 threads = 2 waves; each wave owns 32 nodes (two 16-row WMMA
// subtiles). Weights are read straight from global (128 KB total, L2/L0
// resident on a 192 MB L2). q output is pre-multiplied by vln_weight.
// ---------------------------------------------------------------------------
__global__ __launch_bounds__(64) void node_mlp_kernel(
    const float* __restrict__ x,
    const float* __restrict__ ln_scale, const float* __restrict__ ln_bias,
    const float* __restrict__ vln,
    const float* __restrict__ pw1, const float* __restrict__ pb1,
    const float* __restrict__ pw2, const float* __restrict__ pb2,
    const float* __restrict__ qw1, const float* __restrict__ qb1,
    const float* __restrict__ qw2, const float* __restrict__ qb2,
    float* __restrict__ P, float* __restrict__ Q, int N)
{
    __shared__ float xn[2][32][CH];     // 16 KB: per-wave x_norm tile
    __shared__ float hb[2][16][CH2];    // 16 KB: per-wave hidden tile

    const int wave = threadIdx.x >> 5;
    const int lane = threadIdx.x & 31;
    const int half = lane >> 4;         // 0: lanes 0-15, 1: lanes 16-31
    const int nl   = lane & 15;
    const int ko   = half << 1;         // K-offset of this lane half (0 or 2)
    const int waveNodeBase = blockIdx.x * 64 + wave * 32;

    // ---- LayerNorm: one row per lane --------------------------------------
    {
        const int node = waveNodeBase + lane;
        if (node < N) {
            const float* xr = x + (size_t)node * CH;
            float s = 0.f;
            #pragma unroll
            for (int c = 0; c < CH; ++c) s += xr[c];
            const float mean = s * (1.0f / CH);
            float v = 0.f;
            #pragma unroll
            for (int c = 0; c < CH; ++c) { float d = xr[c] - mean; v += d * d; }
            const float inv = rsqrtf(v * (1.0f / CH) + LN_EPS);
            #pragma unroll
            for (int c = 0; c < CH; ++c)
                xn[wave][lane][c] = (xr[c] - mean) * inv * ln_scale[c] + ln_bias[c];
        } else {
            #pragma unroll
            for (int c = 0; c < CH; ++c) xn[wave][lane][c] = 0.f;
        }
    }
    __syncthreads();

    const float* W1[2] = {pw1, qw1};
    const float* B1[2] = {pb1, qb1};
    const float* W2[2] = {pw2, qw2};
    const float* B2[2] = {pb2, qb2};
    float*       OUT[2] = {P, Q};

    const v8f vzero = {};

    for (int m = 0; m < 2; ++m) {           // 0 = p MLP, 1 = q MLP
        for (int mt = 0; mt < 2; ++mt) {    // two 16-row subtiles of 32 nodes
            // ---- GEMM1: [16 x 64] @ [64 x 128] -> h --------------------------
            v8f acc[8];
            #pragma unroll
            for (int t = 0; t < 8; ++t) acc[t] = vzero;

            for (int k0 = 0; k0 < CH; k0 += 4) {
                v2f a;                       // A frag: row nl, K = k0+ko, k0+ko+1
                a.x = xn[wave][mt * 16 + nl][k0 + ko];
                a.y = xn[wave][mt * 16 + nl][k0 + ko + 1];
                #pragma unroll
                for (int t = 0; t < 8; ++t) {
                    v2f b;                   // B frag: col t*16+nl, matching K split
                    b.x = W1[m][(size_t)(k0 + ko)     * CH2 + (t << 4) + nl];
                    b.y = W1[m][(size_t)(k0 + ko + 1) * CH2 + (t << 4) + nl];
                    acc[t] = __builtin_amdgcn_wmma_f32_16x16x4_f32(
                        false, a, false, b, (short)0, acc[t], false, false);
                }
            }

            // ---- bias + SiLU -> LDS h ---------------------------------------
            #pragma unroll
            for (int t = 0; t < 8; ++t) {
                const int nc = (t << 4) + nl;
                const float bb = B1[m][nc];
                #pragma unroll
                for (int r = 0; r < 8; ++r)
                    hb[wave][r + (half << 3)][nc] = silu_f(acc[t][r] + bb);
            }
            __syncthreads();

            // ---- GEMM2: [16 x 128] @ [128 x 64] -----------------------------
            v8f acc2[4];
            #pragma unroll
            for (int t = 0; t < 4; ++t) acc2[t] = vzero;

            for (int k0 = 0; k0 < CH2; k0 += 4) {
                v2f a;
                a.x = hb[wave][nl][k0 + ko];
                a.y = hb[wave][nl][k0 + ko + 1];
                #pragma unroll
                for (int t = 0; t < 4; ++t) {
                    v2f b;
                    b.x = W2[m][(size_t)(k0 + ko)     * CH + (t << 4) + nl];
                    b.y = W2[m][(size_t)(k0 + ko + 1) * CH + (t << 4) + nl];
                    acc2[t] = __builtin_amdgcn_wmma_f32_16x16x4_f32(
                        false, a, false, b, (short)0, acc2[t], false, false);
                }
            }

            // ---- bias (+vln fold for q) and store ---------------------------
            const int tileBase = waveNodeBase + mt * 16;
            #pragma unroll
            for (int t = 0; t < 4; ++t) {
                const int nc = (t << 4) + nl;
                const float bb = B2[m][nc];
                const float sc = (m == 1) ? vln[nc] : 1.0f;
                #pragma unroll
                for (int r = 0; r < 8; ++r) {
                    const int node = tileBase + r + (half << 3);
                    if (node < N)
                        OUT[m][(size_t)node * CH + nc] = (acc2[t][r] + bb) * sc;
                }
            }
            __syncthreads();
        }
    }
}

// ---------------------------------------------------------------------------
// Edge kernel: 64 threads per edge (one per channel). Memory-bound gather /
// 3x3 contraction / atomic scatter-add. ~1.7 GB traffic -> HBM-limited.
// ---------------------------------------------------------------------------
__global__ __launch_bounds__(256) void edge_kernel(
    const float* __restrict__ P, const float* __restrict__ Q,
    const float* __restrict__ vec, const float* __restrict__ im,
    const int* __restrict__ senders, const int* __restrict__ receivers,
    float* __restrict__ out, int E)
{
    const long long idx = (long long)blockIdx.x * blockDim.x + threadIdx.x;
    const int e = (int)(idx >> 6);
    const int c = (int)(idx & 63);
    if (e >= E) return;

    const int j = senders[e];
    const int i = receivers[e];

    const float qv = Q[(size_t)j * CH + c];        // already includes vln fold
    const float* vj = vec + (size_t)j * 3 * CH + c;
    const float s0 = vj[0]      * qv;
    const float s1 = vj[CH]     * qv;
    const float s2 = vj[2 * CH] * qv;

    const float* M = im + (size_t)e * 9;
    const float f0 = M[0] * s0 + M[1] * s1 + M[2] * s2;
    const float f1 = M[3] * s0 + M[4] * s1 + M[5] * s2;
    const float f2 = M[6] * s0 + M[7] * s1 + M[8] * s2;

    const float pv = P[(size_t)i * CH + c];
    float* o = out + (size_t)i * 3 * CH + c;
    atomicAdd(o,          pv * f0);
    atomicAdd(o + CH,     pv * f1);
    atomicAdd(o + 2 * CH, pv * f2);
}

__global__ void zero_kernel(float* __restrict__ out, long long n)
{
    const long long i = (long long)blockIdx.x * blockDim.x + threadIdx.x;
    if (i < n) out[i] = 0.f;
}

extern "C" void kernel_launch(void* const* d_in, const int* in_sizes, int n_in,
                              void* d_out, int out_size, void* d_ws, size_t ws_size,
                              hipStream_t stream)
{
    const float* x        = (const float*)d_in[0];
    const float* vec      = (const float*)d_in[1];
    const float* im       = (const float*)d_in[2];
    const float* ln_scale = (const float*)d_in[3];
    const float* ln_bias  = (const float*)d_in[4];
    const float* vln      = (const float*)d_in[5];
    const float* pw1      = (const float*)d_in[6];
    const float* pb1      = (const float*)d_in[7];
    const float* pw2      = (const float*)d_in[8];
    const float* pb2      = (const float*)d_in[9];
    const float* qw1      = (const float*)d_in[10];
    const float* qb1      = (const float*)d_in[11];
    const float* qw2      = (const float*)d_in[12];
    const float* qb2      = (const float*)d_in[13];
    const int*   senders  = (const int*)d_in[14];
    const int*   receivers= (const int*)d_in[15];

    const int N = in_sizes[0] / CH;
    const int E = in_sizes[14];

    float* out = (float*)d_out;
    float* P = (float*)d_ws;              // [N, 64]
    float* Q = P + (size_t)N * CH;        // [N, 64] (vln-folded)

    // 1) zero output (harness poisons it; edge kernel accumulates atomically)
    {
        const long long n = (long long)out_size;
        zero_kernel<<<(int)((n + 255) / 256), 256, 0, stream>>>(out, n);
    }

    // 2) LayerNorm + p/q MLPs via WMMA (fp32 16x16x4)
    node_mlp_kernel<<<(N + 63) / 64, 64, 0, stream>>>(
        x, ln_scale, ln_bias, vln,
        pw1, pb1, pw2, pb2, qw1, qb1, qw2, qb2, P, Q, N);

    // 3) edge gather / 3x3 contraction / atomic scatter
    {
        const long long threads = (long long)E * CH;
        edge_kernel<<<(int)((threads + 255) / 256), 256, 0, stream>>>(
            P, Q, vec, im, senders, receivers, out, E);
    }
}